// Upsample_89275190215183
// MI455X (gfx1250) — compile-verified
//
#include <hip/hip_runtime.h>

typedef __attribute__((ext_vector_type(2))) float v2f;
typedef __attribute__((ext_vector_type(8))) float v8f;

#define N_PTS 4096
#define C_CH  64
#define NQ    (3 * N_PTS)          // 12288 query points
#define OUT_COLS (4 * N_PTS)       // 16384

__constant__ const float kHX = 0.0005f;  // SPACING/2
__constant__ const float kHY = 0.0005f;

// One wave handles a 16-query tile; loops over all 256 candidate tiles of 16.
// Score(m,n) = ||c_n||^2 - 2 q_m . c_n, computed as a 16x16x4 f32 WMMA:
//   A[m] = (-2qx, -2qy, 1, 0)   B[:,n] = (cx, cy, cx^2+cy^2, 0)
__global__ __launch_bounds__(128) void nn_argmin_wmma(
    const float* __restrict__ coords,   // [N_PTS, 2]
    int* __restrict__ idx_out)          // [NQ]
{
    const int lane = threadIdx.x & 31;
    const int wave = blockIdx.x * 4 + (threadIdx.x >> 5);
    const int m0   = wave * 16;         // first query of this tile
    const int half = lane >> 4;         // 0: lanes 0-15, 1: lanes 16-31
    const int l15  = lane & 15;

    const float2* c2 = (const float2*)coords;

    // ---- A fragment (16x4 f32): lanes 0-15 carry K=0,1 ; lanes 16-31 K=2,3
    v2f a;
    if (half == 0) {
        int q = m0 + l15;
        int g = q >> 12;                // which of the 3 shifted copies (N=4096)
        int i = q & (N_PTS - 1);
        float dx = (g == 0) ? -kHX : kHX;
        float dy = (g == 1) ? -kHY : kHY;
        float2 p = c2[i];
        a.x = -2.0f * (p.x + dx);
        a.y = -2.0f * (p.y + dy);
    } else {
        a.x = 1.0f;                     // K=2 row of A (picks up ||c||^2)
        a.y = 0.0f;                     // K=3 row of A (padding)
    }

    float best[8];
    int   bidx[8];
#pragma unroll
    for (int v = 0; v < 8; ++v) { best[v] = 3.0e38f; bidx[v] = 0; }

#pragma unroll 2
    for (int t = 0; t < N_PTS / 16; ++t) {
        int n = t * 16 + l15;
        float2 p = c2[n];
        // B fragment (4x16 f32): lanes 0-15 carry K=0,1 ; lanes 16-31 K=2,3
        v2f b;
        if (half == 0) { b.x = p.x; b.y = p.y; }
        else           { b.x = p.x * p.x + p.y * p.y; b.y = 0.0f; }

        v8f c = {};
        // (neg_a, A, neg_b, B, c_mod, C, reuse_a, reuse_b)
        v8f d = __builtin_amdgcn_wmma_f32_16x16x4_f32(
            false, a, false, b, (short)0, c, false, false);

        // D layout: lane<16 -> N=lane, M=v ; lane>=16 -> N=lane-16, M=v+8
#pragma unroll
        for (int v = 0; v < 8; ++v) {
            float s = d[v];
            bool take = s < best[v];    // strict < keeps first occurrence (n ascending)
            best[v] = take ? s : best[v];
            bidx[v] = take ? n : bidx[v];
        }
    }

    // Branchless argmin reduction across the 16 lanes of each half-wave.
#pragma unroll
    for (int off = 1; off < 16; off <<= 1) {
#pragma unroll
        for (int v = 0; v < 8; ++v) {
            float os = __shfl_xor(best[v], off, 32);
            int   oi = __shfl_xor(bidx[v], off, 32);
            // non-short-circuit: force cmp+cndmask, no exec branching
            int take = (int)(os < best[v]) | ((int)(os == best[v]) & (int)(oi < bidx[v]));
            best[v] = take ? os : best[v];
            bidx[v] = take ? oi : bidx[v];
        }
    }

    if (l15 == 0) {
#pragma unroll
        for (int v = 0; v < 8; ++v)
            idx_out[m0 + v + half * 8] = bidx[v];
    }
}

// out[c, 0:4096]      = values[c, :]
// out[c, 4096:16384]  = values[c, idx[col-4096]]
__global__ __launch_bounds__(256) void emit_output(
    const float* __restrict__ values,   // [C_CH, N_PTS]
    const int*   __restrict__ idx,      // [NQ]
    float* __restrict__ out)            // [C_CH, OUT_COLS]
{
    int j   = blockIdx.x * 256 + threadIdx.x;
    int c   = j >> 14;                  // / 16384
    int col = j & (OUT_COLS - 1);
    int src = (col < N_PTS) ? col : idx[col - N_PTS];
    out[j] = values[c * N_PTS + src];
}

extern "C" void kernel_launch(void* const* d_in, const int* in_sizes, int n_in,
                              void* d_out, int out_size, void* d_ws, size_t ws_size,
                              hipStream_t stream) {
    const float* values = (const float*)d_in[0];  // [64, 4096]
    const float* coords = (const float*)d_in[1];  // [4096, 2]
    float* out = (float*)d_out;                   // [64, 16384]
    int*   idx = (int*)d_ws;                      // 12288 ints (48 KB scratch)

    // 768 query tiles of 16; 4 waves (128 threads) per block -> 192 blocks.
    nn_argmin_wmma<<<NQ / 16 / 4, 128, 0, stream>>>(coords, idx);

    // 64 * 16384 outputs, 256 threads/block -> 4096 blocks.
    emit_output<<<(C_CH * OUT_COLS) / 256, 256, 0, stream>>>(values, idx, out);
}